// LHC_64561948393858
// MI455X (gfx1250) — compile-verified
//
#include <hip/hip_runtime.h>
#include <hip/hip_bf16.h>

typedef float v2f __attribute__((ext_vector_type(2)));
typedef float v8f __attribute__((ext_vector_type(8)));

// D = A(16x4) * B(4x16) + C, fp32 WMMA (CDNA5)
#define WMMA_F32X4(A, B, C) \
  __builtin_amdgcn_wmma_f32_16x16x4_f32(false, (A), false, (B), (short)0, (C), false, false)

constexpr int kB = 8, kC = 32, kW = 48, kH = 48;
constexpr int kN = kW * kH;          // 2304 particles / grid cells
constexpr int kNF = 15;              // frames
constexpr int kTiles = kN / 16;      // 144 16-wide tiles
constexpr int kPSplit = 4;           // waves cooperating per cell tile (splat)
constexpr int kPChunk = kTiles / kPSplit;  // 36 particle tiles per wave
constexpr float kPosClamp = 1.1f;
constexpr float kTrunc = 0.8f;
constexpr float kHitD2 = 0.44628710262841953f;   // -2*ln(0.8)
constexpr float kExp2Scale = -0.7213475204444817f;  // -0.5*log2(e)

// ---------------------------------------------------------------------------
// Init: x_part = x_feat, pos = uniform grid
// ---------------------------------------------------------------------------
__global__ __launch_bounds__(256) void k_init(const float* __restrict__ x_feat,
                                              float* __restrict__ x_part,
                                              float* __restrict__ pos) {
  const int stride = gridDim.x * blockDim.x;
  for (int i = blockIdx.x * blockDim.x + threadIdx.x; i < kB * kC * kN; i += stride)
    x_part[i] = x_feat[i];
  for (int i = blockIdx.x * blockDim.x + threadIdx.x; i < kB * kN; i += stride) {
    int b = i / kN, n = i % kN;
    int w = n / kH, h = n % kH;
    pos[(b * 2 + 0) * kN + n] = -1.0f + (2.0f / (kW - 1)) * (float)w;
    pos[(b * 2 + 1) * kN + n] = -1.0f + (2.0f / (kH - 1)) * (float)h;
  }
}

// ---------------------------------------------------------------------------
// Particle update: residual rule-MLP + velocity MLP + pos integrate/clamp.
// One wave32 per (batch, 16-particle tile). All GEMMs via v_wmma_f32_16x16x4_f32.
// ---------------------------------------------------------------------------
__global__ __launch_bounds__(256) void k_mlp(float* __restrict__ x_part,
                                             float* __restrict__ pos,
                                             const float* __restrict__ rw1,
                                             const float* __restrict__ rb1,
                                             const float* __restrict__ rw2,
                                             const float* __restrict__ rb2,
                                             const float* __restrict__ vw1,
                                             const float* __restrict__ vb1,
                                             const float* __restrict__ vw2,
                                             const float* __restrict__ vb2) {
  __shared__ float smem[8][kC * 16];  // per-wave 32x16 tile staging (layout transpose)
  const int lane   = threadIdx.x & 31;
  const int wslot  = threadIdx.x >> 5;
  const int wid    = blockIdx.x * 8 + wslot;        // 0..1151, exact
  const int b      = wid / kTiles;
  const int tile   = wid % kTiles;
  const int n0     = tile * 16;
  const int lane_n = lane & 15;                     // N (or M) index within tile
  const int half   = lane >> 4;                     // 0: K=0,1 / M=r ; 1: K=2,3 / M=r+8
  float* sm = smem[wslot];

  const int xbase = b * kC * kN + n0 + lane_n;      // column base into x_part

  // ---- rule MLP layer 1: h(16xN) = relu(rw1(16x32) @ x(32xN) + rb1) ----
  v8f acc = {};
#pragma unroll
  for (int kb2 = 0; kb2 < 32; kb2 += 4) {
    const int ka = kb2 + 2 * half;
    v2f a, bb;
    a.x  = rw1[lane_n * 32 + ka];
    a.y  = rw1[lane_n * 32 + ka + 1];
    bb.x = x_part[xbase + ka * kN];
    bb.y = x_part[xbase + (ka + 1) * kN];
    acc = WMMA_F32X4(a, bb, acc);
  }
#pragma unroll
  for (int r = 0; r < 8; ++r) {
    const int m = r + 8 * half;
    sm[m * 16 + lane_n] = fmaxf(acc[r] + rb1[m], 0.0f);  // h in (m,n) layout
  }

  // ---- rule MLP layer 2: y(32xN) = relu(rw2(32x16) @ h + rb2); x += y ----
  v2f bch[4];
#pragma unroll
  for (int i = 0; i < 4; ++i) {
    const int ka = i * 4 + 2 * half;
    bch[i].x = sm[ka * 16 + lane_n];
    bch[i].y = sm[(ka + 1) * 16 + lane_n];
  }
  float xn[16];
#pragma unroll
  for (int mt = 0; mt < 2; ++mt) {
    v8f acc2 = {};
#pragma unroll
    for (int i = 0; i < 4; ++i) {
      const int ka = i * 4 + 2 * half;
      v2f a;
      a.x = rw2[(mt * 16 + lane_n) * 16 + ka];
      a.y = rw2[(mt * 16 + lane_n) * 16 + ka + 1];
      acc2 = WMMA_F32X4(a, bch[i], acc2);
    }
#pragma unroll
    for (int r = 0; r < 8; ++r) {
      const int m = mt * 16 + r + 8 * half;
      const float y = fmaxf(acc2[r] + rb2[m], 0.0f);
      xn[mt * 8 + r] = x_part[xbase + m * kN] + y;
    }
  }
#pragma unroll
  for (int mt = 0; mt < 2; ++mt)
#pragma unroll
    for (int r = 0; r < 8; ++r) {
      const int m = mt * 16 + r + 8 * half;
      x_part[xbase + m * kN] = xn[mt * 8 + r];
      sm[m * 16 + lane_n]    = xn[mt * 8 + r];      // x_new in (m,n) layout
    }

  // ---- velocity MLP layer 1: vh = relu(vw1(16x32) @ x_new + vb1) ----
  v8f acc3 = {};
#pragma unroll
  for (int kb2 = 0; kb2 < 32; kb2 += 4) {
    const int ka = kb2 + 2 * half;
    v2f a, bb;
    a.x  = vw1[lane_n * 32 + ka];
    a.y  = vw1[lane_n * 32 + ka + 1];
    bb.x = sm[ka * 16 + lane_n];
    bb.y = sm[(ka + 1) * 16 + lane_n];
    acc3 = WMMA_F32X4(a, bb, acc3);
  }
  float vh[8];
#pragma unroll
  for (int r = 0; r < 8; ++r)
    vh[r] = fmaxf(acc3[r] + vb1[r + 8 * half], 0.0f);

  // ---- velocity head (2x16): per-lane dot + cross-half fold, tanh ----
  float s0 = 0.0f, s1 = 0.0f;
#pragma unroll
  for (int r = 0; r < 8; ++r) {
    const int m = r + 8 * half;
    s0 += vw2[m] * vh[r];
    s1 += vw2[16 + m] * vh[r];
  }
  s0 += __shfl_xor(s0, 16, 32);
  s1 += __shfl_xor(s1, 16, 32);
  const float v0 = tanhf(vb2[0] + s0);
  const float v1 = tanhf(vb2[1] + s1);

  if (half == 0) {
    const int pn = b * 2 * kN + n0 + lane_n;
    float px = pos[pn] + v0;
    float py = pos[pn + kN] + v1;
    pos[pn]      = fminf(fmaxf(px, -kPosClamp), kPosClamp);
    pos[pn + kN] = fminf(fmaxf(py, -kPosClamp), kPosClamp);
  }
}

// ---------------------------------------------------------------------------
// Truncated-exp splat. The WMMA computes the full squared distance tile:
//   A row m = [-2px, -2py, |p|^2, 1],  B col n = [gx, gy, 1, |g|^2]
//   => D[m][n] = |p_m - g_n|^2   (K=4 fully used, no post-fixup / shuffles)
// 4 waves cooperate per 16-cell tile (36 particle tiles each); partial
// ksum/hits combined with LDS ds_add_f32 atomics; block-wide fused write-out.
// Grid: 576 blocks x 8 waves = 1152 cell tiles x 4 splits.
// ---------------------------------------------------------------------------
__global__ __launch_bounds__(256) void k_splat(const float* __restrict__ x_part,
                                               const float* __restrict__ pos,
                                               float* __restrict__ out, int t) {
  __shared__ float s_ks[2][16];
  __shared__ float s_ht[2][16];
  if (threadIdx.x < 32) {
    s_ks[threadIdx.x >> 4][threadIdx.x & 15] = 0.0f;
    s_ht[threadIdx.x >> 4][threadIdx.x & 15] = 0.0f;
  }
  __syncthreads();

  const int lane   = threadIdx.x & 31;
  const int wslot  = threadIdx.x >> 5;
  const int group  = wslot >> 2;           // which of the block's 2 cell tiles
  const int sub    = wslot & 3;            // particle-range split 0..3
  const int ctg    = blockIdx.x * 2 + group;
  const int b      = ctg / kTiles;
  const int ct     = ctg % kTiles;
  const int lane_n = lane & 15;
  const int half   = lane >> 4;
  const int ncell  = ct * 16 + lane_n;

  const int w = ncell / kH, h = ncell % kH;
  const float gx  = -1.0f + (2.0f / (kW - 1)) * (float)w;
  const float gy  = -1.0f + (2.0f / (kH - 1)) * (float)h;

  // B operand (4x16): col n = [gx, gy, 1, |g|^2] — constant over the loop
  v2f bb;
  bb.x = half ? 1.0f : gx;
  bb.y = half ? (gx * gx + gy * gy) : gy;

  const float* posx = pos + b * 2 * kN;
  const float* posy = posx + kN;

  float ks = 0.0f, ht = 0.0f;
  for (int pt = sub * kPChunk; pt < (sub + 1) * kPChunk; ++pt) {
    const int m = pt * 16 + lane_n;        // both halves load the same particle
    const float px = posx[m];
    const float py = posy[m];
    v2f a;                                  // A (16x4): row m = [-2px, -2py, |p|^2, 1]
    a.x = half ? (px * px + py * py) : (-2.0f * px);
    a.y = half ? 1.0f : (-2.0f * py);
    v8f d = {};
    d = WMMA_F32X4(a, bb, d);               // d[r] = |p_{r+8*half} - g_{lane_n}|^2
#pragma unroll
    for (int r = 0; r < 8; ++r) {
      const float dist = d[r];
      const float kval = (dist <= kTrunc) ? exp2f(kExp2Scale * dist) : 0.0f;
      ks += kval;
      ht += (dist < kHitD2) ? 1.0f : 0.0f;  // kval > 0.8 <=> dist < -2 ln 0.8
    }
  }
  // fold the two lane halves (M=0..7 with M=8..15), then combine the 4 subs
  ks += __shfl_xor(ks, 16, 32);
  ht += __shfl_xor(ht, 16, 32);
  if (half == 0) {
    atomicAdd(&s_ks[group][lane_n], ks);    // ds_add_f32
    atomicAdd(&s_ht[group][lane_n], ht);
  }
  __syncthreads();

  // fused frame write-out: 2 tiles x 32 ch x 16 cells = 1024 elems, 4/thread
#pragma unroll
  for (int i = 0; i < 4; ++i) {
    const int idx = threadIdx.x + i * 256;
    const int g   = idx >> 9;
    const int rem = idx & 511;
    const int c   = rem >> 4;
    const int ni  = rem & 15;
    const int cg  = blockIdx.x * 2 + g;
    const int b2  = cg / kTiles;
    const int n2  = (cg % kTiles) * 16 + ni;
    const float scale = s_ks[g][ni] / fmaxf(s_ht[g][ni], 1.0f);
    out[(((size_t)b2 * kNF + t) * kC + c) * kN + n2] =
        x_part[(b2 * kC + c) * kN + n2] * scale;
  }
}

// ---------------------------------------------------------------------------
extern "C" void kernel_launch(void* const* d_in, const int* in_sizes, int n_in,
                              void* d_out, int out_size, void* d_ws, size_t ws_size,
                              hipStream_t stream) {
  const float* x_feat = (const float*)d_in[0];
  const float* rw1 = (const float*)d_in[1];
  const float* rb1 = (const float*)d_in[2];
  const float* rw2 = (const float*)d_in[3];
  const float* rb2 = (const float*)d_in[4];
  const float* vw1 = (const float*)d_in[5];
  const float* vb1 = (const float*)d_in[6];
  const float* vw2 = (const float*)d_in[7];
  const float* vb2 = (const float*)d_in[8];
  float* out = (float*)d_out;

  float* x_part = (float*)d_ws;                   // (B, C, N)  = 2.36 MB
  float* pos    = x_part + (size_t)kB * kC * kN;  // (B, 2, N) = 147 KB

  k_init<<<256, 256, 0, stream>>>(x_feat, x_part, pos);

  const int nblk_mlp   = (kB * kTiles) / 8;            // 144
  const int nblk_splat = (kB * kTiles * kPSplit) / 8;  // 576
  for (int t = 0; t < kNF; ++t) {
    k_mlp<<<nblk_mlp, 256, 0, stream>>>(x_part, pos, rw1, rb1, rw2, rb2,
                                        vw1, vb1, vw2, vb2);
    k_splat<<<nblk_splat, 256, 0, stream>>>(x_part, pos, out, t);
  }
}